// LurieNet_k_5617817223336
// MI455X (gfx1250) — compile-verified
//
#include <hip/hip_runtime.h>
#include <math.h>

// ---------------------------------------------------------------------------
// Problem constants (from the reference)
// ---------------------------------------------------------------------------
#define N128   128
#define MM     (N128 * N128)
#define TMAXC  512
#define BSZ    512
#define KTOP   2
#define STEPF  0.01f
#define EPSF   1e-5f

// Recurrence-kernel tiling
#define BT     32          // batch columns per workgroup (512/32 = 16 WGs)
#define LDA    132         // LDS ld for A/B/C rows (16B-aligned rows, bank-spread)
#define LDN    132         // LDS ld for column-major X/T buffers (16B-aligned, bank-spread)

typedef __attribute__((ext_vector_type(2))) float v2f;
typedef __attribute__((ext_vector_type(8))) float v8f;

// ---------------------------------------------------------------------------
// Setup kernels (one-time cost, plain fp32 — the WMMA path is the recurrence)
// ---------------------------------------------------------------------------

// M = (triu(Z,1) - triu(Z,1)^T) * scale
__global__ void skew_kernel(const float* __restrict__ Z, float* __restrict__ M, float scale) {
    int idx = blockIdx.x * 256 + threadIdx.x;        // 16384 elements
    int r = idx >> 7, c = idx & 127;
    float v = 0.0f;
    if (c > r)      v =  Z[r * N128 + c];
    else if (r > c) v = -Z[c * N128 + r];
    M[idx] = v * scale;
}

// R = I + alpha * M
__global__ void axpyI_kernel(const float* __restrict__ M, float* __restrict__ R, float alpha) {
    int idx = blockIdx.x * 256 + threadIdx.x;
    int r = idx >> 7, c = idx & 127;
    R[idx] = alpha * M[idx] + ((r == c) ? 1.0f : 0.0f);
}

// D = alpha * (A @ op(B)) + (addI ? I : 0);  op(B) = B or B^T
__global__ void gemm128_kernel(const float* __restrict__ A, const float* __restrict__ B,
                               float* __restrict__ D, float alpha, int transB, int addI) {
    int c = blockIdx.x * 16 + threadIdx.x;
    int r = blockIdx.y * 16 + threadIdx.y;
    float s = 0.0f;
    for (int k = 0; k < N128; ++k) {
        float a = A[r * N128 + k];
        float b = transB ? B[c * N128 + k] : B[k * N128 + c];
        s = fmaf(a, b, s);
    }
    D[r * N128 + c] = alpha * s + ((addI && r == c) ? 1.0f : 0.0f);
}

// W[:,j] = U[:,j] * d_j ; mode 0: d_j = |S[j,j]| ; mode 1: d_j = diagArr[j]
__global__ void colscale_kernel(const float* __restrict__ U, const float* __restrict__ S,
                                const float* __restrict__ diagArr, float* __restrict__ W, int mode) {
    int idx = blockIdx.x * 256 + threadIdx.x;
    int c = idx & 127;
    float d = (mode == 0) ? fabsf(S[c * (N128 + 1)]) : diagArr[c];
    W[idx] = U[idx] * d;
}

// A = 0.5*G + 0.5*skew(YA)
__global__ void finalizeA_kernel(const float* __restrict__ G, const float* __restrict__ YA,
                                 float* __restrict__ A) {
    int idx = blockIdx.x * 256 + threadIdx.x;
    int r = idx >> 7, c = idx & 127;
    float sk = 0.0f;
    if (c > r)      sk =  YA[r * N128 + c];
    else if (r > c) sk = -YA[c * N128 + r];
    A[idx] = 0.5f * G[idx] + 0.5f * sk;
}

// Scalar pipeline: top-2 singular values, alpha_upp, SA diagonal (trivial work)
__global__ void sadiag_kernel(const float* __restrict__ SB, const float* __restrict__ SC,
                              const float* __restrict__ GA_ks1, const float* __restrict__ GA_k,
                              const float* __restrict__ GA_kp1, float* __restrict__ SAdiag) {
    if (threadIdx.x != 0 || blockIdx.x != 0) return;
    float b1 = -1.0f, b2 = -1.0f, c1 = -1.0f, c2 = -1.0f;
    for (int i = 0; i < N128; ++i) {
        float vb = fabsf(SB[i * (N128 + 1)]);
        if (vb > b1) { b2 = b1; b1 = vb; } else if (vb > b2) { b2 = vb; }
        float vc = fabsf(SC[i * (N128 + 1)]);
        if (vc > c1) { c2 = c1; c1 = vc; } else if (vc > c2) { c2 = vc; }
    }
    // alpha_upp = sqrt(4*K*G^2 * sum(sing_B^2 * sing_C^2)), K=2, G=1
    float alpha = sqrtf(4.0f * 2.0f * (b1 * b1 * c1 * c1 + b2 * b2 * c2 * c2));
    float ga  = GA_ks1[0];                       // SA1 (K-1 = 1x1)
    float ga2 = fabsf(GA_k[0]) + EPSF;           // GA2
    float sa2 = -(alpha + ga) - ga2;             // SA2
    float mn  = fminf(fminf(ga, sa2), 0.0f);     // jnp.min over 2x2 block-diag incl. zeros
    SAdiag[0] = ga;
    SAdiag[1] = sa2;
    for (int i = 2; i < N128; ++i)
        SAdiag[i] = mn - fabsf(GA_kp1[(i - 2) * (N128 - KTOP) + (i - 2)]);
}

// ---------------------------------------------------------------------------
// Main recurrence kernel: fp32 WMMA (V_WMMA_F32_16X16X4_F32), all-LDS-resident.
// X and T kept column-major (Xt[col][n], ld=LDN) so every B-fragment is one
// contiguous ds_load_b64 and every epilogue store is a contiguous float2.
// ---------------------------------------------------------------------------
__global__ __launch_bounds__(256, 1) void lurie_recurrence(
    const float* __restrict__ X0, const float* __restrict__ Amat,
    const float* __restrict__ Bmat, const float* __restrict__ Cmat,
    const float* __restrict__ bx, const float* __restrict__ by,
    float* __restrict__ out) {
    extern __shared__ float smem[];
    float* sA  = smem;                        // 128 x LDA   (row-major)
    float* sB  = sA  + N128 * LDA;
    float* sC  = sB  + N128 * LDA;
    float* sbx = sC  + N128 * LDA;            // 128
    float* sby = sbx + N128;                  // 128
    float* sXa = sby + N128;                  // BT x LDN  (state ping, column-major)
    float* sXb = sXa + BT * LDN;              // BT x LDN  (state pong)
    float* sT  = sXb + BT * LDN;              // BT x LDN  (tanh buffer)

    const int tid   = threadIdx.x;
    const int bbase = blockIdx.x * BT;

    // Stage A/B/C into LDS (row-padded, float4 vector copies: 32 f4 per row)
    for (int i = tid; i < N128 * 32; i += 256) {
        int row = i >> 5, c4 = i & 31;
        ((float4*)(sA + row * LDA))[c4] = ((const float4*)(Amat + row * N128))[c4];
        ((float4*)(sB + row * LDA))[c4] = ((const float4*)(Bmat + row * N128))[c4];
        ((float4*)(sC + row * LDA))[c4] = ((const float4*)(Cmat + row * N128))[c4];
    }
    if (tid < N128) { sbx[tid] = bx[tid]; sby[tid] = by[tid]; }

    // Stage X0 tile (X0 row (b,:) == Xt column-major row: no transpose needed)
    // and emit the t=0 output slice, fully vectorized.
    for (int i = tid; i < (N128 * BT) / 4; i += 256) {    // 1024 float4
        int bl = i >> 5;
        int n4 = (i & 31) * 4;
        float4 vv = *(const float4*)(X0 + (size_t)(bbase + bl) * N128 + n4);
        *(float4*)(sXa + bl * LDN + n4) = vv;
        *(float4*)(&out[((size_t)(bbase + bl) * TMAXC + 0) * N128 + n4]) = vv;
    }
    __syncthreads();

    const int wave = tid >> 5;
    const int lane = tid & 31;
    const int lh   = lane >> 4;    // half-wave: K pair select / M offset 8
    const int ll   = lane & 15;    // M (A-frag) or N (B/C/D-frag)
    const int m0   = wave * 16;    // this wave's output row block
    const int Mb   = m0 + 8 * lh;  // base accumulator row for this lane

    float* Xc = sXa;
    float* Xn = sXb;

    for (int t = 1; t < TMAXC; ++t) {
        // ---- pass A: T = tanh(C @ Xc + by) ------------------------------
        {
            v8f acc0a = {0.f,0.f,0.f,0.f,0.f,0.f,0.f,0.f};
            v8f acc0b = {0.f,0.f,0.f,0.f,0.f,0.f,0.f,0.f};
            v8f acc1a = {0.f,0.f,0.f,0.f,0.f,0.f,0.f,0.f};
            v8f acc1b = {0.f,0.f,0.f,0.f,0.f,0.f,0.f,0.f};
#pragma unroll 4
            for (int k0 = 0; k0 < N128; k0 += 8) {
                int ar = (m0 + ll) * LDA + k0 + 2 * lh;
                v2f a0 = *(const v2f*)(sC + ar);
                v2f a1 = *(const v2f*)(sC + ar + 4);
                int x0o = ll * LDN + k0 + 2 * lh;            // tile0: col = ll
                int x1o = (16 + ll) * LDN + k0 + 2 * lh;     // tile1: col = 16+ll
                v2f b00 = *(const v2f*)(Xc + x0o);
                v2f b10 = *(const v2f*)(Xc + x1o);
                v2f b01 = *(const v2f*)(Xc + x0o + 4);
                v2f b11 = *(const v2f*)(Xc + x1o + 4);
                acc0a = __builtin_amdgcn_wmma_f32_16x16x4_f32(false, a0, false, b00, (short)0, acc0a, false, false);
                acc1a = __builtin_amdgcn_wmma_f32_16x16x4_f32(false, a0, false, b10, (short)0, acc1a, false, false);
                acc0b = __builtin_amdgcn_wmma_f32_16x16x4_f32(false, a1, false, b01, (short)0, acc0b, false, false);
                acc1b = __builtin_amdgcn_wmma_f32_16x16x4_f32(false, a1, false, b11, (short)0, acc1b, false, false);
            }
            float* T0 = sT + ll * LDN + Mb;
            float* T1 = sT + (16 + ll) * LDN + Mb;
#pragma unroll
            for (int v = 0; v < 8; v += 2) {
                float by0 = sby[Mb + v], by1 = sby[Mb + v + 1];
                float2 t0v, t1v;
                t0v.x = tanhf(acc0a[v]   + acc0b[v]   + by0);
                t0v.y = tanhf(acc0a[v+1] + acc0b[v+1] + by1);
                t1v.x = tanhf(acc1a[v]   + acc1b[v]   + by0);
                t1v.y = tanhf(acc1a[v+1] + acc1b[v+1] + by1);
                *(float2*)(T0 + v) = t0v;
                *(float2*)(T1 + v) = t1v;
            }
        }
        __syncthreads();
        // ---- pass B: Xn = Xc + STEP*(A@Xc + B@T + bx) -------------------
        {
            v8f acc0a = {0.f,0.f,0.f,0.f,0.f,0.f,0.f,0.f};
            v8f acc0b = {0.f,0.f,0.f,0.f,0.f,0.f,0.f,0.f};
            v8f acc1a = {0.f,0.f,0.f,0.f,0.f,0.f,0.f,0.f};
            v8f acc1b = {0.f,0.f,0.f,0.f,0.f,0.f,0.f,0.f};
#pragma unroll 4
            for (int k0 = 0; k0 < N128; k0 += 8) {
                int ar = (m0 + ll) * LDA + k0 + 2 * lh;
                v2f a0 = *(const v2f*)(sA + ar);
                v2f a1 = *(const v2f*)(sA + ar + 4);
                int x0o = ll * LDN + k0 + 2 * lh;
                int x1o = (16 + ll) * LDN + k0 + 2 * lh;
                v2f b00 = *(const v2f*)(Xc + x0o);
                v2f b10 = *(const v2f*)(Xc + x1o);
                v2f b01 = *(const v2f*)(Xc + x0o + 4);
                v2f b11 = *(const v2f*)(Xc + x1o + 4);
                acc0a = __builtin_amdgcn_wmma_f32_16x16x4_f32(false, a0, false, b00, (short)0, acc0a, false, false);
                acc1a = __builtin_amdgcn_wmma_f32_16x16x4_f32(false, a0, false, b10, (short)0, acc1a, false, false);
                acc0b = __builtin_amdgcn_wmma_f32_16x16x4_f32(false, a1, false, b01, (short)0, acc0b, false, false);
                acc1b = __builtin_amdgcn_wmma_f32_16x16x4_f32(false, a1, false, b11, (short)0, acc1b, false, false);
            }
#pragma unroll 4
            for (int k0 = 0; k0 < N128; k0 += 8) {
                int ar = (m0 + ll) * LDA + k0 + 2 * lh;
                v2f a0 = *(const v2f*)(sB + ar);
                v2f a1 = *(const v2f*)(sB + ar + 4);
                int t0o = ll * LDN + k0 + 2 * lh;
                int t1o = (16 + ll) * LDN + k0 + 2 * lh;
                v2f b00 = *(const v2f*)(sT + t0o);
                v2f b10 = *(const v2f*)(sT + t1o);
                v2f b01 = *(const v2f*)(sT + t0o + 4);
                v2f b11 = *(const v2f*)(sT + t1o + 4);
                acc0a = __builtin_amdgcn_wmma_f32_16x16x4_f32(false, a0, false, b00, (short)0, acc0a, false, false);
                acc1a = __builtin_amdgcn_wmma_f32_16x16x4_f32(false, a0, false, b10, (short)0, acc1a, false, false);
                acc0b = __builtin_amdgcn_wmma_f32_16x16x4_f32(false, a1, false, b01, (short)0, acc0b, false, false);
                acc1b = __builtin_amdgcn_wmma_f32_16x16x4_f32(false, a1, false, b11, (short)0, acc1b, false, false);
            }
            const float* Xc0 = Xc + ll * LDN + Mb;
            const float* Xc1 = Xc + (16 + ll) * LDN + Mb;
            float* Xn0 = Xn + ll * LDN + Mb;
            float* Xn1 = Xn + (16 + ll) * LDN + Mb;
#pragma unroll
            for (int v = 0; v < 8; v += 2) {
                float bx0 = sbx[Mb + v], bx1 = sbx[Mb + v + 1];
                float2 xc0 = *(const float2*)(Xc0 + v);
                float2 xc1 = *(const float2*)(Xc1 + v);
                float2 xn0, xn1;
                xn0.x = xc0.x + STEPF * (acc0a[v]   + acc0b[v]   + bx0);
                xn0.y = xc0.y + STEPF * (acc0a[v+1] + acc0b[v+1] + bx1);
                xn1.x = xc1.x + STEPF * (acc1a[v]   + acc1b[v]   + bx0);
                xn1.y = xc1.y + STEPF * (acc1a[v+1] + acc1b[v+1] + bx1);
                *(float2*)(Xn0 + v) = xn0;
                *(float2*)(Xn1 + v) = xn1;
            }
        }
        __syncthreads();
        // ---- stream Xn to HBM: b128 LDS reads -> coalesced float4 stores -
        for (int i = tid; i < (N128 * BT) / 4; i += 256) {
            int bl = i >> 5;
            int n4 = (i & 31) * 4;
            float4 vv = *(const float4*)(Xn + bl * LDN + n4);
            *(float4*)(&out[((size_t)(bbase + bl) * TMAXC + t) * N128 + n4]) = vv;
        }
        float* tmp = Xc; Xc = Xn; Xn = tmp;   // next write to old Xc is behind a barrier
    }
}

// ---------------------------------------------------------------------------
// Host orchestration
// ---------------------------------------------------------------------------
extern "C" void kernel_launch(void* const* d_in, const int* in_sizes, int n_in,
                              void* d_out, int out_size, void* d_ws, size_t ws_size,
                              hipStream_t stream) {
    (void)in_sizes; (void)n_in; (void)out_size; (void)ws_size;

    const float* X0     = (const float*)d_in[0];
    const float* GA_ks1 = (const float*)d_in[1];
    const float* GA_k   = (const float*)d_in[2];
    const float* GA_kp1 = (const float*)d_in[3];
    const float* YA     = (const float*)d_in[4];
    const float* UA     = (const float*)d_in[5];
    const float* UB     = (const float*)d_in[6];
    const float* VB     = (const float*)d_in[7];
    const float* SB     = (const float*)d_in[8];
    const float* UC     = (const float*)d_in[9];
    const float* VC     = (const float*)d_in[10];
    const float* SC     = (const float*)d_in[11];
    const float* bx     = (const float*)d_in[12];
    const float* by     = (const float*)d_in[13];
    float* out = (float*)d_out;

    float* ws = (float*)d_ws;
    float* bufSkew = ws + 0 * MM;
    float* t0      = ws + 1 * MM;
    float* t1      = ws + 2 * MM;
    float* oUA     = ws + 3 * MM;
    float* oUB     = ws + 4 * MM;
    float* oVB     = ws + 5 * MM;
    float* oUC     = ws + 6 * MM;
    float* oVC     = ws + 7 * MM;
    float* Wt      = ws + 8 * MM;
    float* Am      = ws + 9 * MM;
    float* Bm      = ws + 10 * MM;
    float* Cm      = ws + 11 * MM;
    float* SAdiag  = ws + 12 * MM;

    dim3 gG(8, 8), gB(16, 16);

    // expm(skew(Z)) via scaling (1/64) + degree-10 Taylor Horner + 6 squarings
    auto expm = [&](const float* Z, float* dst) {
        skew_kernel<<<64, 256, 0, stream>>>(Z, bufSkew, 1.0f / 64.0f);
        axpyI_kernel<<<64, 256, 0, stream>>>(bufSkew, t0, 1.0f / 10.0f);
        float* cur = t0; float* nxt = t1;
        for (int i = 9; i >= 1; --i) {
            gemm128_kernel<<<gG, gB, 0, stream>>>(bufSkew, cur, nxt, 1.0f / (float)i, 0, 1);
            float* tmp = cur; cur = nxt; nxt = tmp;
        }
        for (int j = 0; j < 5; ++j) {
            gemm128_kernel<<<gG, gB, 0, stream>>>(cur, cur, nxt, 1.0f, 0, 0);
            float* tmp = cur; cur = nxt; nxt = tmp;
        }
        gemm128_kernel<<<gG, gB, 0, stream>>>(cur, cur, dst, 1.0f, 0, 0);
    };

    // SA diagonal (needs only raw diag entries; independent of the expm's)
    sadiag_kernel<<<1, 32, 0, stream>>>(SB, SC, GA_ks1, GA_k, GA_kp1, SAdiag);

    // C = orth(UC) @ diag(|diag SC|) @ orth(VC)^T
    expm(UC, oUC);
    expm(VC, oVC);
    colscale_kernel<<<64, 256, 0, stream>>>(oUC, SC, SAdiag, Wt, 0);
    gemm128_kernel<<<gG, gB, 0, stream>>>(Wt, oVC, Cm, 1.0f, 1, 0);

    // B = orth(UB) @ diag(|diag SB|) @ orth(VB)^T
    expm(UB, oUB);
    expm(VB, oVB);
    colscale_kernel<<<64, 256, 0, stream>>>(oUB, SB, SAdiag, Wt, 0);
    gemm128_kernel<<<gG, gB, 0, stream>>>(Wt, oVB, Bm, 1.0f, 1, 0);

    // A = 0.5 * orth(UA) @ diag(SAdiag) @ orth(UA)^T + 0.5 * skew(YA)
    expm(UA, oUA);
    colscale_kernel<<<64, 256, 0, stream>>>(oUA, SB, SAdiag, Wt, 1);
    gemm128_kernel<<<gG, gB, 0, stream>>>(Wt, oUA, t0, 1.0f, 1, 0);
    finalizeA_kernel<<<64, 256, 0, stream>>>(t0, YA, Am);

    // Main recurrence: 16 WGs x 256 threads, ~254 KB dynamic LDS each
    const size_t smemBytes = (size_t)(3 * N128 * LDA + 2 * N128 + 3 * BT * LDN) * sizeof(float);
    (void)hipFuncSetAttribute((const void*)lurie_recurrence,
                              hipFuncAttributeMaxDynamicSharedMemorySize, (int)smemBytes);
    lurie_recurrence<<<dim3(BSZ / BT), dim3(256), smemBytes, stream>>>(
        X0, Am, Bm, Cm, bx, by, out);
}